// APRMaxPool_31920196943919
// MI455X (gfx1250) — compile-verified
//
#include <hip/hip_runtime.h>

// FILL = -(FLT_MAX / 2), exactly representable (exponent decrement of FLT_MAX).
static constexpr float FILL = -(3.4028234663852886e+38f / 2.0f);

// ---------------------------------------------------------------------------
// Kernel 1: initialize pooled output to FILL with 128-bit stores.
// Runs every launch (harness poisons d_out before timing).
// ---------------------------------------------------------------------------
__global__ void aprmax_fill_kernel(float* __restrict__ out, int n) {
    int n4 = n >> 2;
    int i  = blockIdx.x * blockDim.x + threadIdx.x;
    if (i < n4) {
        float4 v;
        v.x = FILL; v.y = FILL; v.z = FILL; v.w = FILL;
        reinterpret_cast<float4*>(out)[i] = v;     // global_store_b128
    }
    if (i == 0) {                                   // scalar tail (n % 4 elems)
        for (int t = n4 << 2; t < n; ++t) out[t] = FILL;
    }
}

// ---------------------------------------------------------------------------
// Kernel 2: scatter-max. One thread per particle. For each of the 64 (b,c)
// planes: NT load of the intensity (stream data stays out of L2, preserving
// the 67 MB atomic working set), then a fire-and-forget
// GLOBAL_ATOMIC_MAX_NUM_F32 (no-return -> STOREcnt only; s_endpgm's implicit
// wait-idle covers completion).
//
// Loads are batched BATCH at a time ahead of their atomics, and the asm has
// no "memory" clobber, so the compiler keeps BATCH loads in flight instead of
// serializing load->wait->atomic per plane.
// ---------------------------------------------------------------------------
__global__ void aprmax_scatter_kernel(const float* __restrict__ x,
                                      const long long* __restrict__ idx,
                                      float* __restrict__ out,
                                      int n_in, int n_out, int planes) {
    int i = blockIdx.x * blockDim.x + threadIdx.x;
    if (i >= n_in) return;

    // pool_index is int64; values < 2^31 so low dword is the segment id.
    int seg = (int)__builtin_nontemporal_load(&idx[i]);   // th:TH_LOAD_NT

    const float* xp = x + i;      // coalesced across lanes within a plane
    float*       op = out + seg;  // scattered; resolves at L2

    constexpr int BATCH = 16;
    int full = planes / BATCH;

    for (int t = 0; t < full; ++t) {
        float v[BATCH];
        #pragma unroll
        for (int u = 0; u < BATCH; ++u) {                 // BATCH loads issued
            v[u] = __builtin_nontemporal_load(xp);        // back-to-back
            xp += n_in;
        }
        #pragma unroll
        for (int u = 0; u < BATCH; ++u) {
            // CDNA5 native IEEE maximumNumber float atomic, no-return form.
            // No "memory" clobber: ordering vs. the value load is enforced by
            // the data dependency; 'out' is never read in this kernel.
            asm volatile("global_atomic_max_num_f32 %0, %1, off"
                         :
                         : "v"(op), "v"(v[u]));
            op += n_out;
        }
    }
    // Remainder planes (planes % BATCH), for generality.
    for (int p = full * BATCH; p < planes; ++p) {
        float v = __builtin_nontemporal_load(xp);
        xp += n_in;
        asm volatile("global_atomic_max_num_f32 %0, %1, off"
                     :
                     : "v"(op), "v"(v));
        op += n_out;
    }
}

// ---------------------------------------------------------------------------
// Launch wrapper
//   d_in[0] = intensities  [B, C, N_in]  float32   (in_sizes[0] = B*C*N_in)
//   d_in[1] = pool_index   [N_in]        int64     (in_sizes[1] = N_in)
//   d_in[2] = n_out scalar (unused; derived from out_size)
//   d_out   = [B, C, n_out] float32
// ---------------------------------------------------------------------------
extern "C" void kernel_launch(void* const* d_in, const int* in_sizes, int n_in_cnt,
                              void* d_out, int out_size, void* d_ws, size_t ws_size,
                              hipStream_t stream) {
    (void)n_in_cnt; (void)d_ws; (void)ws_size;

    const float*     x   = (const float*)d_in[0];
    const long long* idx = (const long long*)d_in[1];
    float*           out = (float*)d_out;

    const int n_in   = in_sizes[1];               // 2,097,152 particles
    const int planes = in_sizes[0] / n_in;        // B*C = 64
    const int n_out  = out_size / planes;         // 262,144 slots per plane

    // Pass 1: fill output with FILL (float4 stores).
    {
        int n4     = (out_size + 3) / 4;
        int blocks = (n4 + 255) / 256;
        aprmax_fill_kernel<<<blocks, 256, 0, stream>>>(out, out_size);
    }

    // Pass 2: scatter-max with native float max atomics.
    {
        int blocks = (n_in + 255) / 256;          // 8192 blocks of 8 waves
        aprmax_scatter_kernel<<<blocks, 256, 0, stream>>>(x, idx, out,
                                                          n_in, n_out, planes);
    }
}